// GraphAttentionLayer_65077344469188
// MI455X (gfx1250) — compile-verified
//
#include <hip/hip_runtime.h>
#include <hip/hip_bf16.h>

typedef float v2f __attribute__((ext_vector_type(2)));
typedef float v8f __attribute__((ext_vector_type(8)));

#define LRELU(x) ((x) > 0.f ? (x) : 0.3f * (x))

constexpr int Bb     = 32;
constexpr int Nn     = 1024;
constexpr int FIN    = 768;
constexpr int FOUT   = 32;
constexpr int KCHUNK = 64;   // K-chunk staged to LDS per wave
constexpr int APITCH = 68;   // padded LDS row pitch (floats/ints), spreads banks

// CDNA5 async memory->LDS copy (ASYNCcnt path). VDST = LDS byte address
// (flat shared pointer truncated to addr[31:0] per ISA aperture rules),
// VADDR = 64-bit global address.
__device__ __forceinline__ void async_ld_b128(unsigned lds_addr,
                                              const void* gptr) {
  asm volatile("global_load_async_to_lds_b128 %0, %1, off"
               :: "v"(lds_addr), "v"(gptr) : "memory");
}
__device__ __forceinline__ void wait_async0() {
  asm volatile("s_wait_asynccnt 0x0" ::: "memory");
}

// ---------------------------------------------------------------------------
// Kernel 1: h = feat @ W (fp32 WMMA 16x16x4), plus s1 = h.a1, s2 = h.a2.
// 128 threads / 4 waves; each wave owns a 32x32 h tile (two 16-row subtiles
// sharing every B fragment -> 4 WMMAs per pair of W loads, halved L2 traffic).
// ---------------------------------------------------------------------------
__global__ __launch_bounds__(128) void gat_h_s_kernel(
    const float* __restrict__ feat, const float* __restrict__ W,
    const float* __restrict__ a, float* __restrict__ h,
    float* __restrict__ s1, float* __restrict__ s2)
{
  __shared__ float ldsA[4][32 * APITCH];

  const int lane  = threadIdx.x & 31;
  const int wave  = threadIdx.x >> 5;
  const int g0    = blockIdx.x * 128 + wave * 32;  // first global row of tile
  const int rsel  = lane & 15;
  const int hioff = (lane >> 4) << 1;              // K sub-offset: 0 or 2

  v8f c00 = {}, c01 = {};   // rows 0..15  x cols [0..15 | 16..31]
  v8f c10 = {}, c11 = {};   // rows 16..31 x cols [0..15 | 16..31]

  for (int ch = 0; ch < FIN / KCHUNK; ++ch) {
    // async-stage 32x64 fp32 A tile (256B per row, memory->LDS, no VGPR trip)
    #pragma unroll
    for (int it = 0; it < 16; ++it) {
      const int idx = it * 32 + lane;
      const int row = idx >> 4, c4 = idx & 15;
      async_ld_b128((unsigned)(size_t)&ldsA[wave][row * APITCH + c4 * 4],
                    &feat[(size_t)(g0 + row) * FIN + ch * KCHUNK + c4 * 4]);
    }
    wait_async0();
    __syncthreads();   // uniform trip count across waves

    #pragma unroll
    for (int ks = 0; ks < KCHUNK / 4; ++ks) {
      const int k0 = ch * KCHUNK + ks * 4;
      // A 16x4: lanes 0-15 hold K={0,1}, lanes 16-31 K={2,3}, row = lane&15
      v2f A0 = *(const v2f*)&ldsA[wave][rsel * APITCH + ks * 4 + hioff];
      v2f A1 = *(const v2f*)&ldsA[wave][(16 + rsel) * APITCH + ks * 4 + hioff];
      // B 4x16: mirrored K-halves across lane halves, col = lane&15
      const float* wr = &W[(size_t)(k0 + hioff) * FOUT];
      v2f B0, B1;
      B0.x = wr[rsel];        B0.y = wr[FOUT + rsel];
      B1.x = wr[16 + rsel];   B1.y = wr[FOUT + 16 + rsel];
      c00 = __builtin_amdgcn_wmma_f32_16x16x4_f32(false, A0, false, B0,
                                                  (short)0, c00, false, false);
      c01 = __builtin_amdgcn_wmma_f32_16x16x4_f32(false, A0, false, B1,
                                                  (short)0, c01, false, false);
      c10 = __builtin_amdgcn_wmma_f32_16x16x4_f32(false, A1, false, B0,
                                                  (short)0, c10, false, false);
      c11 = __builtin_amdgcn_wmma_f32_16x16x4_f32(false, A1, false, B1,
                                                  (short)0, c11, false, false);
    }
    __syncthreads();
  }

  // epilogue: store h; reduce s1/s2 per row across the 16-lane halves
  const float a1_0 = a[rsel],        a1_1 = a[16 + rsel];
  const float a2_0 = a[FOUT + rsel], a2_1 = a[FOUT + 16 + rsel];
  const int rowhi = (lane >> 4) * 8;   // C layout: upper lanes hold rows M+8

  #pragma unroll
  for (int r = 0; r < 8; ++r) {
    // tile 0 (rows g0 + 0..15)
    {
      const int grow = g0 + r + rowhi;
      float s1p = c00[r] * a1_0 + c01[r] * a1_1;
      float s2p = c00[r] * a2_0 + c01[r] * a2_1;
      #pragma unroll
      for (int m = 1; m <= 8; m <<= 1) {
        s1p += __shfl_xor(s1p, m);
        s2p += __shfl_xor(s2p, m);
      }
      if (rsel == 0) { s1[grow] = s1p; s2[grow] = s2p; }
      h[(size_t)grow * FOUT + rsel]      = c00[r];
      h[(size_t)grow * FOUT + 16 + rsel] = c01[r];
    }
    // tile 1 (rows g0 + 16..31)
    {
      const int grow = g0 + 16 + r + rowhi;
      float s1p = c10[r] * a1_0 + c11[r] * a1_1;
      float s2p = c10[r] * a2_0 + c11[r] * a2_1;
      #pragma unroll
      for (int m = 1; m <= 8; m <<= 1) {
        s1p += __shfl_xor(s1p, m);
        s2p += __shfl_xor(s2p, m);
      }
      if (rsel == 0) { s1[grow] = s1p; s2[grow] = s2p; }
      h[(size_t)grow * FOUT + rsel]      = c10[r];
      h[(size_t)grow * FOUT + 16 + rsel] = c11[r];
    }
  }
}

// ---------------------------------------------------------------------------
// Kernel 2: single-pass masked softmax + attn@h (fp32 WMMA) + ELU.
// grid: (N/32, B); 128 threads / 4 waves split the j range; each wave handles
// TWO 16-row tiles so every h B-fragment feeds 4 WMMAs (halved L2 traffic).
// Shift constant c_i = lrelu(s1_i + max_j s2_j): monotone lrelu => valid
// upper bound, exp args <= 0, masked entries contribute exactly 0 (same as
// exp(MASK_VAL - max) in the reference).
// ---------------------------------------------------------------------------
__global__ __launch_bounds__(128) void gat_attn_kernel(
    const int* __restrict__ adj, const float* __restrict__ h,
    const float* __restrict__ s1, const float* __restrict__ s2,
    float* __restrict__ out)
{
  __shared__ float sS2[Nn];
  __shared__ float sS1[32];
  __shared__ float sPmax[128];
  __shared__ float sMax;
  __shared__ float red[32][FOUT + 1];   // padded numerator accumulators
  __shared__ float sDen[32];
  __shared__ int   ldsAdj[4][32 * APITCH];

  const int t    = threadIdx.x;
  const int lane = t & 31, wave = t >> 5;
  const int b    = blockIdx.y;
  const int i0   = blockIdx.x * 32;
  const size_t bN = (size_t)b * Nn;

  // stage s2[b,:] + per-batch max; stage s1 rows; zero reduction buffers
  float pmax = -1e30f;
  #pragma unroll
  for (int q = 0; q < Nn / 128; ++q) {
    const float v = s2[bN + t + q * 128];
    sS2[t + q * 128] = v;
    pmax = fmaxf(pmax, v);
  }
  sPmax[t] = pmax;
  if (t < 32) { sS1[t] = s1[bN + i0 + t]; sDen[t] = 0.f; }
  for (int idx = t; idx < 32 * (FOUT + 1); idx += 128) ((float*)red)[idx] = 0.f;
  __syncthreads();
  if (t == 0) {
    float m = sPmax[0];
    for (int q = 1; q < 128; ++q) m = fmaxf(m, sPmax[q]);
    sMax = m;
  }
  __syncthreads();

  const int   rsel  = lane & 15;
  const int   hioff = (lane >> 4) << 1;
  const float s1r0  = sS1[rsel];
  const float s1r1  = sS1[16 + rsel];
  const float crow0 = LRELU(s1r0 + sMax);
  const float crow1 = LRELU(s1r1 + sMax);

  v8f c00 = {}, c01 = {}, c10 = {}, c11 = {};
  float dsum0 = 0.f, dsum1 = 0.f;

  for (int jc = wave; jc < Nn / KCHUNK; jc += 4) {   // 4 iterations, uniform
    const int j0 = jc * KCHUNK;
    // async-stage 32x64 int32 adj tile (the dominant HBM stream, read once)
    #pragma unroll
    for (int it = 0; it < 16; ++it) {
      const int idx = it * 32 + lane;
      const int row = idx >> 4, c4 = idx & 15;
      async_ld_b128((unsigned)(size_t)&ldsAdj[wave][row * APITCH + c4 * 4],
                    &adj[(bN + i0 + row) * (size_t)Nn + j0 + c4 * 4]);
    }
    wait_async0();
    __syncthreads();

    #pragma unroll
    for (int st = 0; st < 4; ++st) {       // 4 j-subtiles of 16 (K dim)
      #pragma unroll
      for (int ks = 0; ks < 4; ++ks) {     // 4 wmma K-steps of 4
        const int kk = st * 16 + ks * 4 + hioff;
        const int j  = j0 + kk;
        const int2 ad0 = *(const int2*)&ldsAdj[wave][rsel * APITCH + kk];
        const int2 ad1 = *(const int2*)&ldsAdj[wave][(16 + rsel) * APITCH + kk];
        const v2f  s2v = *(const v2f*)&sS2[j];
        const float e00 = LRELU(s1r0 + s2v.x), e01 = LRELU(s1r0 + s2v.y);
        const float e10 = LRELU(s1r1 + s2v.x), e11 = LRELU(s1r1 + s2v.y);
        const float p00 = (ad0.x > 0) ? __expf(e00 - crow0) : 0.f;
        const float p01 = (ad0.y > 0) ? __expf(e01 - crow0) : 0.f;
        const float p10 = (ad1.x > 0) ? __expf(e10 - crow1) : 0.f;
        const float p11 = (ad1.y > 0) ? __expf(e11 - crow1) : 0.f;
        v2f A0; A0.x = p00; A0.y = p01;
        v2f A1; A1.x = p10; A1.y = p11;
        dsum0 += p00 + p01;
        dsum1 += p10 + p11;
        // B 4x16 from h (4 MB total, L2-resident), shared by both row tiles
        const float* hr = &h[(bN + j) * FOUT];
        v2f B0, B1;
        B0.x = hr[rsel];        B0.y = hr[FOUT + rsel];
        B1.x = hr[16 + rsel];   B1.y = hr[FOUT + 16 + rsel];
        c00 = __builtin_amdgcn_wmma_f32_16x16x4_f32(false, A0, false, B0,
                                                    (short)0, c00, false, false);
        c01 = __builtin_amdgcn_wmma_f32_16x16x4_f32(false, A0, false, B1,
                                                    (short)0, c01, false, false);
        c10 = __builtin_amdgcn_wmma_f32_16x16x4_f32(false, A1, false, B0,
                                                    (short)0, c10, false, false);
        c11 = __builtin_amdgcn_wmma_f32_16x16x4_f32(false, A1, false, B1,
                                                    (short)0, c11, false, false);
      }
    }
  }

  // fold the two K-halves of the denominators, combine waves via LDS atomics
  dsum0 += __shfl_xor(dsum0, 16);
  dsum1 += __shfl_xor(dsum1, 16);
  #pragma unroll
  for (int r = 0; r < 8; ++r) {
    const int row = r + (lane >> 4) * 8;   // C layout rows
    atomicAdd(&red[row][rsel],           c00[r]);
    atomicAdd(&red[row][16 + rsel],      c01[r]);
    atomicAdd(&red[16 + row][rsel],      c10[r]);
    atomicAdd(&red[16 + row][16 + rsel], c11[r]);
  }
  if (lane < 16) {
    atomicAdd(&sDen[lane],      dsum0);
    atomicAdd(&sDen[16 + lane], dsum1);
  }
  __syncthreads();

  // finalize: divide, ELU, store (32x32 tile, 8 values per thread)
  #pragma unroll
  for (int q = 0; q < 8; ++q) {
    const int idx = t + q * 128;
    const int row = idx >> 5, col = idx & 31;
    float v = red[row][col] / sDen[row];
    v = (v > 0.f) ? v : (__expf(v) - 1.f);
    out[(bN + i0 + row) * FOUT + col] = v;
  }
}

// ---------------------------------------------------------------------------
extern "C" void kernel_launch(void* const* d_in, const int* in_sizes, int n_in,
                              void* d_out, int out_size, void* d_ws,
                              size_t ws_size, hipStream_t stream) {
  const float* feat = (const float*)d_in[0];
  const int*   adj  = (const int*)d_in[1];
  const float* W    = (const float*)d_in[2];
  const float* a    = (const float*)d_in[3];
  float* out = (float*)d_out;

  float* h  = (float*)d_ws;                    // B*N*FOUT = 4 MB
  float* s1 = h + (size_t)Bb * Nn * FOUT;      // B*N
  float* s2 = s1 + (size_t)Bb * Nn;            // B*N

  gat_h_s_kernel<<<dim3((Bb * Nn) / 128), dim3(128), 0, stream>>>(
      feat, W, a, h, s1, s2);
  gat_attn_kernel<<<dim3(Nn / 32, Bb), dim3(128), 0, stream>>>(
      adj, h, s1, s2, out);
}